// BiDecoder_43069932044947
// MI455X (gfx1250) — compile-verified
//
#include <hip/hip_runtime.h>
#include <hip/hip_bf16.h>

typedef float v2f __attribute__((ext_vector_type(2)));
typedef float v8f __attribute__((ext_vector_type(8)));

// -------------------------------------------------------------------------
// Kernel 1: vh[b] = ifeat @ Ps[b]    ([NM,128] x [128,128] -> [NM,128]), b=0,1
// Math: vh[b][m][d] = sum_e ifeat[m,e] * Ps[b][e,d]
// (equivalent to the reference's user-side transform via
//  ufeat[u]^T Ps[b]^T ifeat[m] == ufeat[u] . (ifeat[m] @ Ps[b]) — item side
//  is 2x cheaper since N_M < N_U)
// One wave computes one 16x16 output tile with V_WMMA_F32_16X16X4_F32,
// accumulating over K=128 in steps of 4 (32 WMMAs per tile).
// -------------------------------------------------------------------------
__global__ void __launch_bounds__(256)
basis_transform_wmma(const float* __restrict__ ifeat,
                     const float* __restrict__ Ps,
                     float* __restrict__ vh,
                     int NM, int mtiles)
{
    const int lane = threadIdx.x & 31;
    const int wave = (blockIdx.x * blockDim.x + threadIdx.x) >> 5;
    const int tiles_per_basis = mtiles * 8;     // 128/16 = 8 N-tiles
    const int b = wave / tiles_per_basis;
    if (b >= 2) return;
    const int t  = wave - b * tiles_per_basis;
    const int mt = t >> 3;
    const int nt = t & 7;

    // ISA layout, 32-bit A 16x4: lanes 0-15 hold K=k+0,k+1; lanes 16-31 hold K=k+2,k+3
    const int rowA = mt * 16 + (lane & 15);
    const int col  = nt * 16 + (lane & 15);
    const int kh   = (lane >> 4) * 2;

    const float* __restrict__ A  = ifeat;                       // [NM,128] row-major (K contiguous)
    const float* __restrict__ Bm = Ps + (size_t)b * 128 * 128;  // [128(K),128(N)] row-major

    const int rA = (rowA < NM) ? rowA : (NM - 1);  // clamp (NM%16==0 for this problem)

    v8f acc = {};
    #pragma unroll 4
    for (int k = 0; k < 128; k += 4) {
        v2f a = *(const v2f*)(A + (size_t)rA * 128 + (k + kh));      // 8B-aligned
        v2f bb;
        bb.x = Bm[(size_t)(k + kh + 0) * 128 + col];
        bb.y = Bm[(size_t)(k + kh + 1) * 128 + col];
        acc = __builtin_amdgcn_wmma_f32_16x16x4_f32(false, a, false, bb,
                                                    (short)0, acc, false, false);
    }

    // D layout: VGPR i holds row (i + 8*(lane>>4)) of the tile, col = lane&15
    float* __restrict__ out = vh + (size_t)b * NM * 128;
    const int rbase = mt * 16 + (lane >> 4) * 8;

    if (mt * 16 + 16 <= NM) {
        // Full tile (the only case when NM % 16 == 0): straight-line stores,
        // no per-row exec-mask juggling.
        float* __restrict__ p = out + (size_t)rbase * 128 + col;
        #pragma unroll
        for (int i = 0; i < 8; ++i) {
            p[(size_t)i * 128] = acc[i];
        }
    } else {
        // Ragged boundary tile (not hit for NM % 16 == 0)
        #pragma unroll
        for (int i = 0; i < 8; ++i) {
            const int r = rbase + i;
            if (r < NM) out[(size_t)r * 128 + col] = acc[i];
        }
    }
}

// -------------------------------------------------------------------------
// Kernel 2: one wave32 per edge.
//   s_b = ufeat[src[e]] . vh[b][dst[e]]   (lane holds a float4 slice, D=128)
//   out[e,c] = s_0 * W[c,0] + s_1 * W[c,1]
// Gather working set (ufeat 51.2MB + vh 51.2MB) fits in the 192MB L2;
// output (40MB) is streamed with non-temporal stores to keep L2 resident.
// -------------------------------------------------------------------------
__global__ void __launch_bounds__(256)
edge_score(const float* __restrict__ ufeat,
           const float* __restrict__ vh,
           const int* __restrict__ src,
           const int* __restrict__ dst,
           const float* __restrict__ W,   // [5,2] row-major
           float* __restrict__ out,       // [E,5]
           int E, int NM)
{
    const int lane = threadIdx.x & 31;
    const int e = (blockIdx.x * blockDim.x + threadIdx.x) >> 5;
    if (e >= E) return;

    const int u = src[e];
    const int m = dst[e];

    const float4 uu = *(const float4*)(ufeat + (size_t)u * 128 + lane * 4);
    const float4 v0 = *(const float4*)(vh    + (size_t)m * 128 + lane * 4);
    const float4 v1 = *(const float4*)(vh + (size_t)NM * 128 + (size_t)m * 128 + lane * 4);

    float s0 = uu.x * v0.x + uu.y * v0.y + uu.z * v0.z + uu.w * v0.w;
    float s1 = uu.x * v1.x + uu.y * v1.y + uu.z * v1.z + uu.w * v1.w;

    // wave32 butterfly reduction — result in all lanes
    #pragma unroll
    for (int off = 16; off > 0; off >>= 1) {
        s0 += __shfl_xor(s0, off, 32);
        s1 += __shfl_xor(s1, off, 32);
    }

    if (lane < 5) {
        const float r = s0 * W[lane * 2 + 0] + s1 * W[lane * 2 + 1];
        __builtin_nontemporal_store(r, out + (size_t)e * 5 + lane);
    }
}

extern "C" void kernel_launch(void* const* d_in, const int* in_sizes, int n_in,
                              void* d_out, int out_size, void* d_ws, size_t ws_size,
                              hipStream_t stream)
{
    const float* ufeat = (const float*)d_in[0];   // [N_U,128]
    const float* ifeat = (const float*)d_in[1];   // [N_M,128]
    const float* Ps    = (const float*)d_in[2];   // [2,128,128]
    const float* W     = (const float*)d_in[3];   // [5,2]
    const int*   src   = (const int*)d_in[4];     // [E]
    const int*   dst   = (const int*)d_in[5];     // [E]
    float* out = (float*)d_out;                   // [E,5]
    float* vh  = (float*)d_ws;                    // scratch: [2, N_M, 128] f32 (51.2 MB)

    const int D  = 128;
    const int NM = in_sizes[1] / D;
    const int E  = in_sizes[4];

    // Phase 1: item-side basis transform (WMMA f32 GEMM)
    const int mtiles  = (NM + 15) / 16;
    const int waves   = 2 * mtiles * 8;
    const int blocks1 = (waves * 32 + 255) / 256;
    basis_transform_wmma<<<blocks1, 256, 0, stream>>>(ifeat, Ps, vh, NM, mtiles);

    // Phase 2: per-edge gather + dual dot + class combine (one wave per edge)
    const long long thr = (long long)E * 32;
    const int blocks2 = (int)((thr + 255) / 256);
    edge_score<<<blocks2, 256, 0, stream>>>(ufeat, vh, src, dst, W, out, E, NM);
}